// Loop_52467320487991
// MI455X (gfx1250) — compile-verified
//
#include <hip/hip_runtime.h>

// ---------------------------------------------------------------------------
// Graves-attention RNN for MI455X (gfx1250).
// Latency-bound serial chain (200 steps); all weights (~21MB) L2-resident.
// Persistent kernel, 40 WGs x 128 thr (4 waves), f32 WMMA 16x16x4 for GEMMs,
// device-scope acquire/release grid barriers (4/step) between stages.
// ---------------------------------------------------------------------------

typedef __attribute__((ext_vector_type(2))) float v2f;
typedef __attribute__((ext_vector_type(8))) float v8f;

#define BB    32
#define SEQL  512
#define DD    320
#define KK    10
#define CC    10
#define DK    3200   // D*K
#define DPP   256
#define DOO   64
#define NWG   40
#define ALIGN_C 0.05f
#define COEF_C  0.3989422917366028f

// workspace layout (float offsets)
#define OFF_SA    0u        // S buffer A (32x3200)
#define OFF_SB    102400u   // S buffer B
#define OFF_NAPAD 204800u   // Na_w padded to 3200x32
#define OFF_FO    307200u   // fo (32x3200)
#define OFF_IDT   409600u   // idt (32x256)
#define OFF_IDENT 417792u   // ident (32x256)
#define OFF_MU    427008u   // mu (32x10)
#define OFF_SIG   427328u   // sig (32x10)
#define OFF_G     427648u   // g (32x10)
#define OFF_O     427968u   // o_{t-1} (32x64)
#define OFF_U     430016u   // u (32x320)
#define OFF_BAR   440256u   // 2 x u32 barrier state

__device__ __forceinline__ void gbar(unsigned* bar) {
  __syncthreads();
  if (threadIdx.x == 0) {
    __threadfence();  // flush our stores to device scope
    unsigned g = __hip_atomic_load(&bar[1], __ATOMIC_RELAXED, __HIP_MEMORY_SCOPE_AGENT);
    unsigned a = __hip_atomic_fetch_add(&bar[0], 1u, __ATOMIC_ACQ_REL, __HIP_MEMORY_SCOPE_AGENT);
    if (a == (unsigned)(NWG - 1)) {
      __hip_atomic_store(&bar[0], 0u, __ATOMIC_RELAXED, __HIP_MEMORY_SCOPE_AGENT);
      __hip_atomic_fetch_add(&bar[1], 1u, __ATOMIC_RELEASE, __HIP_MEMORY_SCOPE_AGENT);
    } else {
      while (__hip_atomic_load(&bar[1], __ATOMIC_ACQUIRE, __HIP_MEMORY_SCOPE_AGENT) == g)
        __builtin_amdgcn_s_sleep(2);
    }
  }
  __syncthreads();  // acquire above invalidates WGP$ -> cross-WG data visible
}

// ---------------------------- init kernels ---------------------------------

__global__ __launch_bounds__(256) void k_init0(const int* __restrict__ spk,
                                               const float* __restrict__ LutS,
                                               float* __restrict__ ws) {
  int tid = blockIdx.x * 256 + threadIdx.x;
  if (tid < BB * DPP) {
    int b = tid >> 8, d = tid & 255;
    ws[OFF_IDENT + tid] = LutS[spk[b] * DPP + d];
  }
  if (tid < BB * DOO) ws[OFF_O + tid] = 0.f;
  if (tid < BB * CC)  ws[OFF_MU + tid] = 0.f;
  if (tid < 2) ((unsigned*)(ws + OFF_BAR))[tid] = 0u;
}

__global__ __launch_bounds__(256) void k_initNa(const float* __restrict__ Na_w,
                                                float* __restrict__ ws) {
  int tid = blockIdx.x * 256 + threadIdx.x;  // 3200*32
  if (tid < DK * 32) {
    int k = tid >> 5, n = tid & 31;
    ws[OFF_NAPAD + tid] = (n < 30) ? Na_w[k * 30 + n] : 0.f;
  }
}

__global__ __launch_bounds__(256) void k_initIdt(const float* __restrict__ Fu_w,
                                                 const float* __restrict__ Fu_b,
                                                 float* __restrict__ ws) {
  int tid = blockIdx.x * 256 + threadIdx.x;  // 32*256
  if (tid < BB * DPP) {
    int b = tid >> 8, d = tid & 255;
    const float* id = ws + OFF_IDENT + b * DPP;
    float acc = Fu_b[d];
    for (int k = 0; k < DPP; ++k) acc += id[k] * Fu_w[k * DPP + d];
    ws[OFF_IDT + tid] = tanhf(acc);
  }
}

__global__ __launch_bounds__(256) void k_initFo(const float* __restrict__ Fo_w,
                                                const float* __restrict__ Fo_b,
                                                float* __restrict__ ws) {
  int tid = blockIdx.x * 256 + threadIdx.x;  // 32*3200
  if (tid < BB * DK) {
    int b = tid / DK, n = tid % DK;
    const float* id = ws + OFF_IDENT + b * DPP;
    float acc = Fo_b[n];
    for (int k = 0; k < DPP; ++k) acc += id[k] * Fo_w[k * DK + n];
    ws[OFF_FO + tid] = acc;
  }
}

__global__ __launch_bounds__(256) void k_initS(float* __restrict__ ws) {
  int tid = blockIdx.x * 256 + threadIdx.x;  // 32*3200
  if (tid < BB * DK) {
    int b = tid / DK, r = tid % DK, d = r / KK;
    ws[OFF_SA + tid] = (d < DPP) ? ws[OFF_IDENT + b * DPP + d] : 0.f;
  }
}

// ------------------------- persistent RNN kernel ---------------------------

__global__ __launch_bounds__(128)
void rnn_persistent(const int* __restrict__ sentence,
                    const float* __restrict__ LutP,
                    const float* __restrict__ Na_b,
                    const float* __restrict__ Nu_w, const float* __restrict__ Nu_b,
                    const float* __restrict__ No_w, const float* __restrict__ No_b,
                    const int* __restrict__ Tptr,
                    float* __restrict__ outp,
                    float* __restrict__ ws) {
  const int wg   = blockIdx.x;
  const int tid  = threadIdx.x;
  const int wid  = tid >> 5;
  const int lane = tid & 31;
  const int lm   = lane & 15;
  const int lh   = lane >> 4;

  float* Sa    = ws + OFF_SA;
  float* Sb    = ws + OFF_SB;
  float* Napad = ws + OFF_NAPAD;
  float* fo    = ws + OFF_FO;
  float* idt   = ws + OFF_IDT;
  float* mu    = ws + OFF_MU;
  float* sig   = ws + OFF_SIG;
  float* gg    = ws + OFF_G;
  float* obuf  = ws + OFF_O;
  float* ubuf  = ws + OFF_U;
  unsigned* bar = (unsigned*)(ws + OFF_BAR);

  __shared__ float s_red[4 * 256];   // cross-wave K reduction / attn tile
  __shared__ float s_alpha[SEQL];
  __shared__ int   s_idx[SEQL];
  __shared__ float s_prm[32];

  const int T = Tptr[0];

  for (int t = 0; t < T; ++t) {
    float* cur = (t & 1) ? Sb : Sa;   // S_{t-1}
    float* nxt = (t & 1) ? Sa : Sb;   // becomes Sp then S_t

    // ---- Stage 1: WG0 = Na GEMM (32x3200 @ 3200x32) + attn params;
    //               others = delay-line shift nxt[:,:,1..9] = cur[:,:,0..8]
    if (wg == 0) {
      int mt = wid >> 1, nt = wid & 1;
      const float* Arow = cur + (mt * 16 + lm) * DK;
      const float* Bw   = Napad + nt * 16 + lm;
      v8f acc = {};
      for (int k = 0; k < DK; k += 4) {
        int ka = k + 2 * lh;
        v2f a, b;
        a.x = Arow[ka];        a.y = Arow[ka + 1];
        b.x = Bw[ka * 32];     b.y = Bw[(ka + 1) * 32];
        acc = __builtin_amdgcn_wmma_f32_16x16x4_f32(false, a, false, b,
                                                    (short)0, acc, false, false);
      }
      int n = nt * 16 + lm;
      float bias = (n < 30) ? Na_b[n] : 0.f;
#pragma unroll
      for (int i = 0; i < 8; ++i)
        s_red[(mt * 16 + i + 8 * lh) * 32 + n] = acc[i] + bias;
      __syncthreads();
      if (tid < BB) {  // per-batch: mu update, sig, softmax(g)
        int b = tid;
        float yv[CC], ev[CC], ymax = -1e30f;
        for (int c = 0; c < CC; ++c) {
          float kt = s_red[b * 32 + c];
          float bt = s_red[b * 32 + 10 + c];
          float yt = s_red[b * 32 + 20 + c];
          mu[b * CC + c] += ALIGN_C * __expf(kt);
          sig[b * CC + c] = __expf(bt);
          yv[c] = yt; ymax = fmaxf(ymax, yt);
        }
        float se = 0.f;
        for (int c = 0; c < CC; ++c) { ev[c] = __expf(yv[c] - ymax); se += ev[c]; }
        float inv = 1.f / se;
        for (int c = 0; c < CC; ++c) gg[b * CC + c] = ev[c] * inv;
      }
    } else {
      for (int e = tid + (wg - 1) * 128; e < BB * DD * 9; e += 128 * (NWG - 1)) {
        int m = e / (DD * 9), r = e % (DD * 9);
        int d = r / 9, j = r % 9;
        nxt[m * DK + d * KK + j + 1] = cur[m * DK + d * KK + j];
      }
    }
    gbar(bar);

    // ---- Stage 2: WGs 0..31 (one batch each): alpha, c_t, Sp slot0
    if (wg < BB) {
      int b = wg;
      if (tid < CC) {
        s_prm[tid]      = mu[b * CC + tid];
        s_prm[10 + tid] = sig[b * CC + tid];
        s_prm[20 + tid] = gg[b * CC + tid];
      }
      for (int s = tid; s < SEQL; s += 128) s_idx[s] = sentence[b * SEQL + s];
      __syncthreads();
      for (int s = tid; s < SEQL; s += 128) {
        float J = (float)(s + 1), acc = 0.f;
#pragma unroll
        for (int c = 0; c < CC; ++c) {
          float dmu = s_prm[c] - J;
          acc += s_prm[20 + c] * __expf(-0.5f * s_prm[10 + c] * dmu * dmu);
        }
        s_alpha[s] = COEF_C * acc;
      }
      __syncthreads();
      for (int d = tid; d < DPP; d += 128) {   // c_t[b,d] = alpha . Lut_P rows
        float acc = 0.f;
        for (int s = 0; s < SEQL; ++s)
          acc += s_alpha[s] * LutP[s_idx[s] * DPP + d];
        nxt[b * DK + d * KK] = acc + idt[b * DPP + d];   // Sp slot0, d<256
      }
      if (tid < DOO)                                      // Sp slot0, d>=256
        nxt[b * DK + (DPP + tid) * KK] = obuf[b * DOO + tid] * (1.0f / 30.0f);
    }
    gbar(bar);

    // ---- Stage 3: Nu GEMM (32x3200 @ 3200x320): 40 tiles, K split 4 waves
    {
      int mt = wg / 20, nt = wg % 20;
      const float* Arow = nxt + (mt * 16 + lm) * DK;
      const float* Bw   = Nu_w + nt * 16 + lm;
      v8f acc = {};
      int k0 = wid * 800, k1 = k0 + 800;
      for (int k = k0; k < k1; k += 4) {
        int ka = k + 2 * lh;
        v2f a, b;
        a.x = Arow[ka];        a.y = Arow[ka + 1];
        b.x = Bw[ka * DD];     b.y = Bw[(ka + 1) * DD];
        acc = __builtin_amdgcn_wmma_f32_16x16x4_f32(false, a, false, b,
                                                    (short)0, acc, false, false);
      }
#pragma unroll
      for (int i = 0; i < 8; ++i) s_red[wid * 256 + lane * 8 + i] = acc[i];
      __syncthreads();
      if (wid == 0) {
#pragma unroll
        for (int i = 0; i < 8; ++i) {
          float v = s_red[lane * 8 + i] + s_red[256 + lane * 8 + i] +
                    s_red[512 + lane * 8 + i] + s_red[768 + lane * 8 + i];
          int m = mt * 16 + i + 8 * lh, n = nt * 16 + lm;
          ubuf[m * DD + n] = v + Nu_b[n];
        }
      }
    }
    gbar(bar);

    // ---- Stage 4: WGs 0..7: scatter u -> S_t slot0 (duplicated, same values)
    //               then No GEMM on A = S_t + fo; write o_t
    if (wg < 8) {
      for (int e = tid; e < BB * DD; e += 128) {
        int m = e / DD, d = e % DD;
        nxt[m * DK + d * KK] = ubuf[e];
      }
      __threadfence();
      __syncthreads();
      int mt = wg >> 2, nt = wg & 3;
      const float* Arow = nxt + (mt * 16 + lm) * DK;
      const float* Frow = fo  + (mt * 16 + lm) * DK;
      const float* Bw   = No_w + nt * 16 + lm;
      v8f acc = {};
      int k0 = wid * 800, k1 = k0 + 800;
      for (int k = k0; k < k1; k += 4) {
        int ka = k + 2 * lh;
        v2f a, b;
        a.x = Arow[ka] + Frow[ka];
        a.y = Arow[ka + 1] + Frow[ka + 1];
        b.x = Bw[ka * DOO];   b.y = Bw[(ka + 1) * DOO];
        acc = __builtin_amdgcn_wmma_f32_16x16x4_f32(false, a, false, b,
                                                    (short)0, acc, false, false);
      }
#pragma unroll
      for (int i = 0; i < 8; ++i) s_red[wid * 256 + lane * 8 + i] = acc[i];
      __syncthreads();
      if (wid == 0) {
#pragma unroll
        for (int i = 0; i < 8; ++i) {
          float v = s_red[lane * 8 + i] + s_red[256 + lane * 8 + i] +
                    s_red[512 + lane * 8 + i] + s_red[768 + lane * 8 + i];
          int m = mt * 16 + i + 8 * lh, n = nt * 16 + lm;
          float o = v + No_b[n];
          obuf[m * DOO + n] = o;
          outp[(size_t)t * (BB * DOO) + m * DOO + n] = o;
        }
      }
    }
    gbar(bar);
  }
}

// ------------------------------- launch ------------------------------------

extern "C" void kernel_launch(void* const* d_in, const int* in_sizes, int n_in,
                              void* d_out, int out_size, void* d_ws, size_t ws_size,
                              hipStream_t stream) {
  (void)in_sizes; (void)n_in; (void)out_size; (void)ws_size;
  const int*   sentence = (const int*)d_in[0];
  const int*   spk      = (const int*)d_in[1];
  const float* LutP     = (const float*)d_in[2];
  const float* LutS     = (const float*)d_in[3];
  const float* Na_w     = (const float*)d_in[4];
  const float* Na_b     = (const float*)d_in[5];
  const float* Nu_w     = (const float*)d_in[6];
  const float* Nu_b     = (const float*)d_in[7];
  const float* No_w     = (const float*)d_in[8];
  const float* No_b     = (const float*)d_in[9];
  const float* Fu_w     = (const float*)d_in[10];
  const float* Fu_b     = (const float*)d_in[11];
  const float* Fo_w     = (const float*)d_in[12];
  const float* Fo_b     = (const float*)d_in[13];
  const int*   Tptr     = (const int*)d_in[14];
  float* ws  = (float*)d_ws;
  float* out = (float*)d_out;

  k_init0 <<<32, 256, 0, stream>>>(spk, LutS, ws);
  k_initNa<<<400, 256, 0, stream>>>(Na_w, ws);
  k_initIdt<<<32, 256, 0, stream>>>(Fu_w, Fu_b, ws);
  k_initFo<<<400, 256, 0, stream>>>(Fo_w, Fo_b, ws);
  k_initS <<<400, 256, 0, stream>>>(ws);
  rnn_persistent<<<NWG, 128, 0, stream>>>(sentence, LutP, Na_b, Nu_w, Nu_b,
                                          No_w, No_b, Tptr, out, ws);
}